// ExponentialMovingAverage_69432441307520
// MI455X (gfx1250) — compile-verified
//
#include <hip/hip_runtime.h>

// EMA scan y_t = w*x_t + (1-w)*y_{t-1} over [B,C,F,T] f32, reformulated as
// chunked 16x16x16 f32 matmuls against the constant decay matrix using
// V_WMMA_F32_16X16X4_F32 (exact f32 precision).

typedef float v2f __attribute__((ext_vector_type(2)));
typedef float v8f __attribute__((ext_vector_type(8)));

__device__ __forceinline__ v8f wmma_f32_16x16x4(v2f a, v2f b, v8f c) {
    // (neg_a, A, neg_b, B, c_mod, C, reuse_a, reuse_b)
    return __builtin_amdgcn_wmma_f32_16x16x4_f32(false, a, false, b, (short)0, c,
                                                 false, false);
}

__global__ __launch_bounds__(32)
void ema_wmma_kernel(const float* __restrict__ x,
                     const float* __restrict__ init,
                     const float* __restrict__ wptr,
                     float* __restrict__ y,
                     int rows, int T)
{
    const int lane = threadIdx.x & 31;
    const int grp  = lane >> 4;   // half-wave: selects K-pair / row-half per ISA layout
    const int l16  = lane & 15;   // A row index / D time-column index
    const int rbase = blockIdx.x * 16;

    // w read on device, clipped like the reference
    float w = wptr[0];
    w = fminf(fmaxf(w, 0.0f), 1.0f);
    const float a = 1.0f - w;

    float apow[17];
    apow[0] = 1.0f;
#pragma unroll
    for (int i = 1; i <= 16; ++i) apow[i] = apow[i - 1] * a;

    // B_j (4x16, K x t): lane l16 = t column; K = 4j + 2*grp + vgpr  (ISA B layout)
    // Mdecay[k][t] = w * a^(t-k) for k <= t else 0
    v2f Bm[4];
#pragma unroll
    for (int j = 0; j < 4; ++j) {
#pragma unroll
        for (int idx = 0; idx < 2; ++idx) {
            int k = 4 * j + 2 * grp + idx;
            Bm[j][idx] = (k <= l16) ? (w * apow[l16 - k]) : 0.0f;
        }
    }

    const float at1 = apow[l16 + 1];   // a^{t+1}, t = l16 (carry weight per column)

    // carry: rows rbase + grp*8 + v  (matches D VGPR layout: M = v + 8*grp)
    v8f carry;
#pragma unroll
    for (int v = 0; v < 8; ++v) {
        int r = rbase + grp * 8 + v;
        carry[v] = (r < rows) ? init[r] : 0.0f;
    }

    // A loads: lane supplies row rbase+l16, contiguous float2 per WMMA step
    int arow = rbase + l16;
    const bool rowok = arow < rows;
    if (arow >= rows) arow = rows - 1;           // clamp (reads only; stores guarded)
    const float* rp = x + (size_t)arow * (size_t)T;
    float*       wp = y + (size_t)(rbase + l16) * (size_t)T;

    // LDS transpose tile [t][r], stride 20 floats -> conflict-free b128/b32 banks
    __shared__ __align__(16) float tile[16 * 20];

    const int koff = 2 * grp;
    const int nchunk = T >> 4;                   // T = 6000 -> 375 chunks

    v2f A[4], An[4];
#pragma unroll
    for (int j = 0; j < 4; ++j) A[j] = *(const v2f*)(rp + 4 * j + koff);
#pragma unroll
    for (int j = 0; j < 4; ++j) An[j] = A[j];

    for (int c = 0; c < nchunk; ++c) {
        const int t0 = c << 4;

        // software pipeline: next chunk's A (independent of carry chain)
        if (c + 1 < nchunk) {
            const float* np = rp + t0 + 16;
#pragma unroll
            for (int j = 0; j < 4; ++j) An[j] = *(const v2f*)(np + 4 * j + koff);
        }
        if (c + 2 < nchunk)
            __builtin_prefetch(rp + t0 + 32 + koff, 0, 0);   // global_prefetch

        // seed accumulator with carry-in term a^{t+1} * carry[row]
        v8f acc;
#pragma unroll
        for (int v = 0; v < 8; ++v) acc[v] = at1 * carry[v];

        // 16x16x16 f32 product as 4 chained 16x16x4 WMMAs
        acc = wmma_f32_16x16x4(A[0], Bm[0], acc);
        acc = wmma_f32_16x16x4(A[1], Bm[1], acc);
        acc = wmma_f32_16x16x4(A[2], Bm[2], acc);
        acc = wmma_f32_16x16x4(A[3], Bm[3], acc);

        // carry-out = column t=15 -> lane 15 (rows 0-7) / lane 31 (rows 8-15)
        const int src = (lane & 16) | 15;
#pragma unroll
        for (int v = 0; v < 8; ++v) carry[v] = __shfl(acc[v], src, 32);

        // stage D tile to LDS: lane holds column t=l16, rows grp*8..grp*8+7
        float* tp = &tile[l16 * 20 + grp * 8];
        float4 s0, s1;
        s0.x = acc[0]; s0.y = acc[1]; s0.z = acc[2]; s0.w = acc[3];
        s1.x = acc[4]; s1.y = acc[5]; s1.z = acc[6]; s1.w = acc[7];
        *(float4*)(tp)     = s0;   // ds_store_b128
        *(float4*)(tp + 4) = s1;
        __syncthreads();

        // transpose read: lane -> row l16, times grp*8 .. grp*8+7
        float o[8];
#pragma unroll
        for (int k = 0; k < 8; ++k) o[k] = tile[(grp * 8 + k) * 20 + l16];
        if (rowok) {
            float4 t0v, t1v;
            t0v.x = o[0]; t0v.y = o[1]; t0v.z = o[2]; t0v.w = o[3];
            t1v.x = o[4]; t1v.y = o[5]; t1v.z = o[6]; t1v.w = o[7];
            float4* dst = (float4*)(wp + t0 + grp * 8);   // 16B-aligned, contiguous
            dst[0] = t0v;                                  // global_store_b128
            dst[1] = t1v;
        }
        __syncthreads();

#pragma unroll
        for (int j = 0; j < 4; ++j) A[j] = An[j];
    }
}

extern "C" void kernel_launch(void* const* d_in, const int* in_sizes, int n_in,
                              void* d_out, int out_size, void* d_ws, size_t ws_size,
                              hipStream_t stream) {
    const float* mag  = (const float*)d_in[0];   // [B,C,F,T] f32
    const float* init = (const float*)d_in[1];   // [B,C,F,1] f32
    const float* wts  = (const float*)d_in[2];   // [1] f32
    float* out = (float*)d_out;                  // [B,C,F,T] f32

    const int rows = in_sizes[1];                // B*C*F = 4112
    const int T    = in_sizes[0] / rows;         // 6000 (multiple of 16)
    const int groups = (rows + 15) / 16;         // 257

    ema_wmma_kernel<<<dim3(groups), dim3(32), 0, stream>>>(mag, init, wts, out,
                                                           rows, T);
}